// Topological_Loss_2989297238291
// MI455X (gfx1250) — compile-verified
//
#include <hip/hip_runtime.h>
#include <math.h>

typedef __attribute__((ext_vector_type(2))) float v2f;
typedef __attribute__((ext_vector_type(8))) float v8f;

#define LAM 0.1f
#define HINF 1e18

// Monotone map f32 -> u32 so unsigned compare == (value, index) lexicographic order.
__device__ __forceinline__ unsigned long long pack_key(float f, int idx) {
  unsigned u = __float_as_uint(f);
  u = (u & 0x80000000u) ? ~u : (u | 0x80000000u);
  return (((unsigned long long)u) << 32) | (unsigned)idx;
}

__device__ __forceinline__ float sigmoidf_(float x) {
  return 1.0f / (1.0f + expf(-x));
}

// Hungarian cost C[i][j] built on the fly (f32 math, cast to f64 like the reference).
__device__ __forceinline__ double hcost(int i, int j, int n, int m,
                                        const float* mB, const float* mD,
                                        const float* pB, const float* pD) {
  if (i < n) {
    if (j < m) {
      float db = fabsf(mB[i] - pB[j]);
      float dd = fabsf(mD[i] - pD[j]);
      float ch = fmaxf(db, dd);
      return (double)(ch * ch);
    }
    if (j - m == i) { float d = (mD[i] - mB[i]) * 0.5f; return (double)(d * d); }
    return 1000000.0;
  } else {
    if (j < m) {
      if (j == i - n) { float d = (pD[j] - pB[j]) * 0.5f; return (double)(d * d); }
      return 1000000.0;
    }
    return 0.0;
  }
}

__global__ __launch_bounds__(256) void topo_loss_kernel(
    const float* __restrict__ model_output,
    const float* __restrict__ labels,
    float* __restrict__ w2_per_sample) {
  __shared__ float Rr[2][16 * 128];          // 8-row partial sums, [0]=pred src, [1]=msk src
  __shared__ float pooled[2][256];           // [0]=pred(16x16, sigmoid), [1]=msk(16x16)
  __shared__ unsigned long long skey[256];   // sort keys (value,index)
  __shared__ int parent[256];
  __shared__ int birthp[256];
  __shared__ float dgm[4][256];              // 0:mask birth,1:mask death,2:pred birth,3:pred death
  __shared__ double hu[512], hv[512], hminv[512];
  __shared__ int hp[512], hway[512], hused[512];
  __shared__ int colArr[512];
  __shared__ double redv[256];
  __shared__ int redj[256];
  __shared__ int s_cnt[2];
  __shared__ int s_j0, s_j1;
  __shared__ double s_delta;

  const int tid = threadIdx.x;
  const int smp = blockIdx.x;
  const float* mo = model_output + (size_t)smp * 16384;
  const float* lb = labels + (size_t)smp * 16384;

  // ---- Phase 1a: 8-row block sums -> Rr[img][16][128] (coalesced over columns) ----
  for (int e = tid; e < 2048; e += 256) {
    int br = e >> 7, c = e & 127;
    const float* b0 = mo + (br << 3) * 128 + c;
    const float* b1 = lb + (br << 3) * 128 + c;
    float s0 = 0.f, s1 = 0.f;
#pragma unroll
    for (int r = 0; r < 8; ++r) { s0 += b0[r * 128]; s1 += b1[r * 128]; }
    Rr[0][e] = s0;
    Rr[1][e] = s1;
  }
  __syncthreads();

  // ---- Phase 1b: column-block reduce as GEMM via V_WMMA_F32_16X16X4_F32 ----
  // pooled(16x16) = Rr(16x128) @ Scol(128x16) * 1/64 ; wave0 -> pred, wave1 -> msk.
  {
    const int wave = tid >> 5;
    const int lane = tid & 31;
    if (wave < 2) {
      const float* Rp = Rr[wave];
      const int ln = lane & 15;               // A: row M ; B/D: col N
      const int koff = (lane < 16) ? 0 : 2;   // K offset for this half-wave
      v8f acc = {0.f, 0.f, 0.f, 0.f, 0.f, 0.f, 0.f, 0.f};
      for (int k0 = 0; k0 < 128; k0 += 4) {
        v2f a, bb;
        a[0] = Rp[ln * 128 + k0 + koff];
        a[1] = Rp[ln * 128 + k0 + koff + 1];
        bb[0] = (((k0 + koff) >> 3) == ln) ? 1.0f : 0.0f;      // Scol[k][N]
        bb[1] = (((k0 + koff + 1) >> 3) == ln) ? 1.0f : 0.0f;
        acc = __builtin_amdgcn_wmma_f32_16x16x4_f32(
            false, a, false, bb, (short)0, acc, false, false);
      }
#pragma unroll
      for (int j = 0; j < 8; ++j) {
        int M = j + ((lane < 16) ? 0 : 8);
        float val = acc[j] * (1.0f / 64.0f);
        pooled[wave][M * 16 + ln] = (wave == 0) ? sigmoidf_(val) : val;
      }
    }
  }
  __syncthreads();

  // ---- Phase 2: PH0 (sublevel, 4-connectivity, elder rule) for msk then pred ----
  for (int which = 0; which < 2; ++which) {
    const float* img = (which == 0) ? pooled[1] : pooled[0];  // mask first
    float* dB = dgm[which * 2 + 0];
    float* dD = dgm[which * 2 + 1];
    skey[tid] = pack_key(img[tid], tid);
    parent[tid] = -1;
    __syncthreads();
    // bitonic sort ascending, 256 elems / 256 threads
    for (int k = 2; k <= 256; k <<= 1) {
      for (int jj = k >> 1; jj > 0; jj >>= 1) {
        int ixj = tid ^ jj;
        if (ixj > tid) {
          bool up = ((tid & k) == 0);
          unsigned long long x = skey[tid], y = skey[ixj];
          if ((x > y) == up) { skey[tid] = y; skey[ixj] = x; }
        }
        __syncthreads();
      }
    }
    if (tid == 0) {
      int cnt = 0;
      for (int t = 0; t < 256; ++t) {
        int idx = (int)(skey[t] & 0xffffffffull);
        parent[idx] = idx;
        birthp[idx] = idx;
        int r = idx >> 4, cc = idx & 15;
        int nb[4]; int nn = 0;
        if (r > 0)   nb[nn++] = idx - 16;
        if (r < 15)  nb[nn++] = idx + 16;
        if (cc > 0)  nb[nn++] = idx - 1;
        if (cc < 15) nb[nn++] = idx + 1;
        for (int q = 0; q < nn; ++q) {
          int j = nb[q];
          if (parent[j] != -1) {
            int ra = idx; while (parent[ra] != ra) { parent[ra] = parent[parent[ra]]; ra = parent[ra]; }
            int rb = j;   while (parent[rb] != rb) { parent[rb] = parent[parent[rb]]; rb = parent[rb]; }
            if (ra != rb) {
              unsigned long long ka = pack_key(img[birthp[ra]], birthp[ra]);
              unsigned long long kb = pack_key(img[birthp[rb]], birthp[rb]);
              int elder, young;
              if (ka <= kb) { elder = ra; young = rb; } else { elder = rb; young = ra; }
              dB[cnt] = img[birthp[young]];
              dD[cnt] = img[idx];
              ++cnt;
              parent[young] = elder;
            }
          }
        }
      }
      s_cnt[which] = cnt;
    }
    __syncthreads();
  }

  // ---- Phase 3: Hungarian (JV potentials, float64) on (n+m)x(n+m) implicit cost ----
  const int n = s_cnt[0], m = s_cnt[1];
  const int N = n + m;
  if (N > 0) {
    for (int j = tid; j <= N; j += 256) { hu[j] = 0.0; hv[j] = 0.0; hp[j] = 0; hway[j] = 0; }
    __syncthreads();
    for (int i = 1; i <= N; ++i) {
      for (int j = tid; j <= N; j += 256) { hminv[j] = HINF; hused[j] = 0; }
      if (tid == 0) { hp[0] = i; s_j0 = 0; }
      __syncthreads();
      while (true) {
        if (tid == 0) hused[s_j0] = 1;
        __syncthreads();
        const int j0 = s_j0;
        const int i0 = hp[j0];
        const double u_i0 = hu[i0];
        double lmin = HINF; int larg = 0x7fffffff;
        for (int j = 1 + tid; j <= N; j += 256) {
          if (!hused[j]) {
            double cur = hcost(i0 - 1, j - 1, n, m, dgm[0], dgm[1], dgm[2], dgm[3]) - u_i0 - hv[j];
            if (cur < hminv[j]) { hminv[j] = cur; hway[j] = j0; }
            double mv = hminv[j];
            if (mv < lmin) { lmin = mv; larg = j; }   // strict < keeps smallest j per thread
          }
        }
        redv[tid] = lmin; redj[tid] = larg;
        __syncthreads();
        for (int st = 128; st > 0; st >>= 1) {        // (min, argmin) with smallest-j tiebreak
          if (tid < st) {
            double ov = redv[tid + st]; int oj = redj[tid + st];
            if (ov < redv[tid] || (ov == redv[tid] && oj < redj[tid])) { redv[tid] = ov; redj[tid] = oj; }
          }
          __syncthreads();
        }
        if (tid == 0) { s_delta = redv[0]; s_j1 = redj[0]; }
        __syncthreads();
        const double delta = s_delta;
        for (int j = tid; j <= N; j += 256) {
          if (hused[j]) { hu[hp[j]] += delta; hv[j] -= delta; }
          else if (j >= 1) hminv[j] -= delta;
        }
        __syncthreads();
        if (tid == 0) s_j0 = s_j1;
        __syncthreads();
        if (hp[s_j0] == 0) break;
      }
      if (tid == 0) {                                  // augment along 'way'
        int j0 = s_j0;
        while (j0) { int j1 = hway[j0]; hp[j0] = hp[j1]; j0 = j1; }
      }
      __syncthreads();
    }
    for (int j = 1 + tid; j <= N; j += 256) colArr[hp[j] - 1] = j - 1;
    __syncthreads();

    // ---- per-sample W2^2 in f32 (matched + diagonals), then sqrt ----
    float lper = 0.f;
    for (int r = tid; r < N; r += 256) {
      int cidx = colArr[r];
      if (r < n) {
        if (cidx < m) {
          float db = fabsf(dgm[0][r] - dgm[2][cidx]);
          float dd = fabsf(dgm[1][r] - dgm[3][cidx]);
          float ch = fmaxf(db, dd);
          lper += ch * ch;
        } else {
          float d = (dgm[1][r] - dgm[0][r]) * 0.5f;
          lper += d * d;
        }
      } else if (cidx < m) {
        float d = (dgm[3][cidx] - dgm[2][cidx]) * 0.5f;
        lper += d * d;
      }
    }
    redv[tid] = (double)lper;
    __syncthreads();
    for (int st = 128; st > 0; st >>= 1) {
      if (tid < st) redv[tid] += redv[tid + st];
      __syncthreads();
    }
    if (tid == 0) {
      float per = (float)redv[0];
      w2_per_sample[smp] = (per > 0.f) ? sqrtf(per) : 0.f;
    }
  } else if (tid == 0) {
    w2_per_sample[smp] = 0.f;
  }
}

// Deterministic final reduction (no float atomics across blocks).
__global__ __launch_bounds__(256) void topo_finalize_kernel(
    const float* __restrict__ w2, float* __restrict__ out, int B) {
  __shared__ float buf[256];
  int tid = threadIdx.x;
  float acc = 0.f;
  for (int i = tid; i < B; i += 256) acc += w2[i];
  buf[tid] = acc;
  __syncthreads();
  for (int st = 128; st > 0; st >>= 1) {
    if (tid < st) buf[tid] += buf[tid + st];
    __syncthreads();
  }
  if (tid == 0) out[0] = LAM * buf[0] / (float)B;
}

extern "C" void kernel_launch(void* const* d_in, const int* in_sizes, int n_in,
                              void* d_out, int out_size, void* d_ws, size_t ws_size,
                              hipStream_t stream) {
  (void)n_in; (void)out_size; (void)ws_size;
  const float* model_output = (const float*)d_in[0];
  const float* labels = (const float*)d_in[1];
  float* out = (float*)d_out;
  float* w2 = (float*)d_ws;                 // B floats of scratch
  const int B = in_sizes[0] / (128 * 128);  // 512

  topo_loss_kernel<<<B, 256, 0, stream>>>(model_output, labels, w2);
  topo_finalize_kernel<<<1, 256, 0, stream>>>(w2, out, B);
}